// GABlock_12618613916381
// MI455X (gfx1250) — compile-verified
//
#include <hip/hip_runtime.h>
#include <math.h>

typedef float v2f __attribute__((ext_vector_type(2)));
typedef float v8f __attribute__((ext_vector_type(8)));

#define LL  512
#define FF  128
#define CC  64
#define HH  12
#define QKD 32
#define VD  32
#define PQ  8
#define PV  8
#define NQK (HH*QKD)    // 384
#define NP3 (HH*PQ*3)   // 288
#define CAT 1440        // H*C + H*Vd + H*Pv*3
#define KT  64          // z rows staged to LDS per async stage

static __device__ __forceinline__ v8f wmma_f32(v2f a, v2f b, v8f c) {
    // D = A(16x4 f32) * B(4x16 f32) + C(16x16 f32)
    return __builtin_amdgcn_wmma_f32_16x16x4_f32(false, a, false, b, (short)0, c, false, false);
}

static __device__ __forceinline__ float wsum(float v) {
#pragma unroll
    for (int o = 16; o; o >>= 1) v += __shfl_xor(v, o, 32);
    return v;
}
static __device__ __forceinline__ float wmax(float v) {
#pragma unroll
    for (int o = 16; o; o >>= 1) v = fmaxf(v, __shfl_xor(v, o, 32));
    return v;
}

// ---------------- K1a: x @ {Wq,Wk,Wv,Wqp,Wkp,Wvp} (one block per row i) ------
__global__ void k_proj(const float* __restrict__ x,
                       const float* __restrict__ Wq, const float* __restrict__ Wk,
                       const float* __restrict__ Wv,
                       const float* __restrict__ Wqp, const float* __restrict__ Wkp,
                       const float* __restrict__ Wvp,
                       float* __restrict__ Q, float* __restrict__ Kb, float* __restrict__ V,
                       float* __restrict__ QPL, float* __restrict__ KPL, float* __restrict__ VPL) {
    __shared__ float xs[FF];
    const int i = blockIdx.x;
    const int t = threadIdx.x;
    if (t < FF) xs[t] = x[i * FF + t];
    __syncthreads();
    for (int c = t; c < NQK; c += blockDim.x) {
        float sq = 0.f, sk = 0.f, sv = 0.f;
        for (int f = 0; f < FF; ++f) {
            const float xv = xs[f];
            sq = fmaf(xv, Wq[f * NQK + c], sq);
            sk = fmaf(xv, Wk[f * NQK + c], sk);
            sv = fmaf(xv, Wv[f * NQK + c], sv);
        }
        Q [i * NQK + c] = sq;
        Kb[i * NQK + c] = sk;
        V [i * NQK + c] = sv;
    }
    for (int c = t; c < NP3; c += blockDim.x) {
        float sq = 0.f, sk = 0.f, sv = 0.f;
        for (int f = 0; f < FF; ++f) {
            const float xv = xs[f];
            sq = fmaf(xv, Wqp[f * NP3 + c], sq);
            sk = fmaf(xv, Wkp[f * NP3 + c], sk);
            sv = fmaf(xv, Wvp[f * NP3 + c], sv);
        }
        QPL[i * NP3 + c] = sq;
        KPL[i * NP3 + c] = sk;
        VPL[i * NP3 + c] = sv;
    }
}

// ---------------- K1b: local -> global point transform (R p + t) -------------
__global__ void k_l2g(const float* __restrict__ R, const float* __restrict__ t,
                      const float* __restrict__ QPL, const float* __restrict__ KPL,
                      const float* __restrict__ VPL,
                      float* __restrict__ QPG, float* __restrict__ KPG,
                      float* __restrict__ VPG) {
    const int NPTS = LL * (NP3 / 3);
    int gid = blockIdx.x * blockDim.x + threadIdx.x;
    if (gid >= 3 * NPTS) return;
    const int a = gid / NPTS;
    const int r = gid % NPTS;
    const int i = r / (NP3 / 3);
    const int p = r % (NP3 / 3);
    const float* src = (a == 0) ? QPL : (a == 1) ? KPL : VPL;
    float*       dst = (a == 0) ? QPG : (a == 1) ? KPG : VPG;
    const float l0 = src[i * NP3 + p * 3 + 0];
    const float l1 = src[i * NP3 + p * 3 + 1];
    const float l2 = src[i * NP3 + p * 3 + 2];
    const float* Ri = R + i * 9;
    dst[i * NP3 + p * 3 + 0] = Ri[0] * l0 + Ri[1] * l1 + Ri[2] * l2 + t[i * 3 + 0];
    dst[i * NP3 + p * 3 + 1] = Ri[3] * l0 + Ri[4] * l1 + Ri[5] * l2 + t[i * 3 + 1];
    dst[i * NP3 + p * 3 + 2] = Ri[6] * l0 + Ri[7] * l1 + Ri[8] * l2 + t[i * 3 + 2];
}

// ---------------- K1c: per-(i,h) squared norms of qp/kp ----------------------
__global__ void k_sumsq(const float* __restrict__ QPG, const float* __restrict__ KPG,
                        float* __restrict__ Q2, float* __restrict__ K2) {
    int gid = blockIdx.x * blockDim.x + threadIdx.x;
    if (gid >= LL * HH) return;
    const int i = gid / HH, h = gid % HH;
    float sq = 0.f, sk = 0.f;
    for (int d = 0; d < PQ * 3; ++d) {
        const float a = QPG[i * NP3 + h * 24 + d];
        const float b = KPG[i * NP3 + h * 24 + d];
        sq = fmaf(a, a, sq);
        sk = fmaf(b, b, sk);
    }
    Q2[gid] = sq;
    K2[gid] = sk;
}

// ---------------- K2a: pair logits  z(262144x64) @ Wpair(64x12) --------------
__global__ void __launch_bounds__(32) k_pair(const float* __restrict__ z,
                                             const float* __restrict__ Wpair,
                                             float* __restrict__ LOG) {
    const int lane = threadIdx.x;
    const int rowbase = blockIdx.x * 16;
    const int m = lane & 15;
    const int koff = (lane >> 4) * 2;
    v2f bf[16];
#pragma unroll
    for (int s = 0; s < 16; ++s) {
        const int kb = s * 4 + koff;
        v2f b = {0.f, 0.f};
        if (m < HH) { b.x = Wpair[kb * HH + m]; b.y = Wpair[(kb + 1) * HH + m]; }
        bf[s] = b;
    }
    const float* zr = z + (size_t)(rowbase + m) * CC;
    v8f acc = {};
#pragma unroll
    for (int s = 0; s < 16; ++s) {
        v2f a = *reinterpret_cast<const v2f*>(zr + s * 4 + koff);
        acc = wmma_f32(a, bf[s], acc);
    }
#pragma unroll
    for (int r = 0; r < 8; ++r) {
        const int mm = r + (lane >> 4) * 8;   // pair-row within tile
        if (m < HH) LOG[(size_t)m * (LL * LL) + rowbase + mm] = acc[r];
    }
}

// ---------------- K2b: node + spatial logits, combine + mask -----------------
__global__ void __launch_bounds__(32) k_logits(const float* __restrict__ Q,
                                               const float* __restrict__ Km,
                                               const float* __restrict__ QPG,
                                               const float* __restrict__ KPG,
                                               const float* __restrict__ Q2,
                                               const float* __restrict__ K2,
                                               const float* __restrict__ coef,
                                               const unsigned char* __restrict__ mask,
                                               float* __restrict__ LOG) {
    const int lane = threadIdx.x;
    const int ib = blockIdx.x * 16, jb = blockIdx.y * 16, h = blockIdx.z;
    const int m = lane & 15;
    const int koff = (lane >> 4) * 2;

    const float* arow = Q  + (size_t)(ib + m) * NQK + h * QKD;
    const float* brow = Km + (size_t)(jb + m) * NQK + h * QKD;
    v8f an = {};
#pragma unroll
    for (int s = 0; s < 8; ++s) {
        v2f a = *reinterpret_cast<const v2f*>(arow + s * 4 + koff);
        v2f b = *reinterpret_cast<const v2f*>(brow + s * 4 + koff);
        an = wmma_f32(a, b, an);
    }
    const float* ap = QPG + (size_t)(ib + m) * NP3 + h * 24;
    const float* bp = KPG + (size_t)(jb + m) * NP3 + h * 24;
    v8f as = {};
#pragma unroll
    for (int s = 0; s < 6; ++s) {
        v2f a = *reinterpret_cast<const v2f*>(ap + s * 4 + koff);
        v2f b = *reinterpret_cast<const v2f*>(bp + s * 4 + koff);
        as = wmma_f32(a, b, as);
    }
    const float gamma = logf(1.f + expf(coef[h]));      // softplus
    const float kc = -gamma * (1.f / 12.f);             // -g*sqrt(2/(9*Pq))/2
    const size_t base = (size_t)h * (LL * LL);
#pragma unroll
    for (int r = 0; r < 8; ++r) {
        const int mm = r + (lane >> 4) * 8;
        const int i = ib + mm, j = jb + m;
        const float node = an[r] * 0.17677669529663687f;   // 1/sqrt(32)
        const float ssd = Q2[i * HH + h] + K2[j * HH + h] - 2.f * as[r];
        float lg = (node + LOG[base + (size_t)i * LL + j] + ssd * kc)
                   * 0.57735026918962576f;                 // sqrt(1/3)
        if (!(mask[i] && mask[j])) lg -= 100000.f;
        LOG[base + (size_t)i * LL + j] = lg;
    }
}

// ---------------- K3: softmax over j, in place -> alpha ----------------------
__global__ void __launch_bounds__(32) k_softmax(const unsigned char* __restrict__ mask,
                                                float* __restrict__ LOG) {
    const int i = blockIdx.x, h = blockIdx.y, lane = threadIdx.x;
    float* row = LOG + (size_t)h * (LL * LL) + (size_t)i * LL;
    float ev[16];
    float mx = -3.0e38f;
#pragma unroll
    for (int s = 0; s < 16; ++s) { const float v = row[lane + s * 32]; ev[s] = v; mx = fmaxf(mx, v); }
    mx = wmax(mx);
    float sum = 0.f;
#pragma unroll
    for (int s = 0; s < 16; ++s) { const float e = expf(ev[s] - mx); ev[s] = e; sum += e; }
    sum = wsum(sum);
    const float inv = mask[i] ? (1.f / sum) : 0.f;
#pragma unroll
    for (int s = 0; s < 16; ++s) row[lane + s * 32] = ev[s] * inv;
}

// ---------------- K4a: feat_node = alpha @ v  (K=512) ------------------------
__global__ void __launch_bounds__(32) k_aggr_node(const float* __restrict__ ALPHA,
                                                  const float* __restrict__ V,
                                                  float* __restrict__ FEAT) {
    const int lane = threadIdx.x;
    const int ib = blockIdx.x * 16, h = blockIdx.y, d0 = blockIdx.z * 16;
    const int m = lane & 15;
    const int koff = (lane >> 4) * 2;
    const float* arow = ALPHA + (size_t)h * (LL * LL) + (size_t)(ib + m) * LL;
    const int off = h * VD + d0 + m;
    v8f acc = {};
    for (int s = 0; s < LL / 4; ++s) {
        const int kb = s * 4 + koff;
        v2f a = *reinterpret_cast<const v2f*>(arow + kb);
        v2f b;
        b.x = V[(size_t)kb * NQK + off];
        b.y = V[(size_t)(kb + 1) * NQK + off];
        acc = wmma_f32(a, b, acc);
    }
#pragma unroll
    for (int r = 0; r < 8; ++r) {
        const int mm = r + (lane >> 4) * 8;
        FEAT[(size_t)(ib + mm) * CAT + (HH * CC) + h * VD + d0 + m] = acc[r];
    }
}

// ------ K4b: feat_p2n = alpha @ z[i]; z streamed via async-to-LDS ------------
// One block (4 waves) per row i. LDS double buffer: 2 stages x (64 rows x 64 c).
// Each wave owns one 16-wide c-tile; z is fetched ONCE per row via
// global_load_async_to_lds_b128 (ASYNCcnt), overlapped with WMMA.
__global__ void __launch_bounds__(128) k_aggr_pair(const float* __restrict__ ALPHA,
                                                   const float* __restrict__ z,
                                                   float* __restrict__ FEAT) {
    __shared__ float zbuf[2][KT * CC];           // 2 x 16 KB
    const int tid  = threadIdx.x;
    const int wave = tid >> 5;
    const int lane = tid & 31;
    const int i    = blockIdx.x;
    const int c0   = wave * 16;
    const int m    = lane & 15;                  // head row (pad to 16) / N col
    const int koff = (lane >> 4) * 2;

    const float* zrow = z + (size_t)i * LL * CC;

    // issue stage 0 (16 KB, contiguous) into zbuf[0]
    {
        const unsigned lbase = (unsigned)(size_t)(&zbuf[0][0]);
        const unsigned long long sbase = (unsigned long long)(size_t)zrow;
#pragma unroll
        for (int n = 0; n < 8; ++n) {
            const unsigned off   = (unsigned)tid * 16u + (unsigned)n * 2048u;
            const unsigned laddr = lbase + off;
            asm volatile("global_load_async_to_lds_b128 %0, %1, %2"
                         :: "v"(laddr), "v"(off), "s"(sbase) : "memory");
        }
    }

    const float* arow = ALPHA + (size_t)m * (LL * LL) + (size_t)i * LL;
    v8f acc = {};
    for (int st = 0; st < LL / KT; ++st) {
        if (st + 1 < LL / KT) {
            // prefetch next stage, then retire current stage's 8 async loads
            const unsigned lbase = (unsigned)(size_t)(&zbuf[(st + 1) & 1][0]);
            const unsigned long long sbase =
                (unsigned long long)(size_t)(zrow + (size_t)(st + 1) * KT * CC);
#pragma unroll
            for (int n = 0; n < 8; ++n) {
                const unsigned off   = (unsigned)tid * 16u + (unsigned)n * 2048u;
                const unsigned laddr = lbase + off;
                asm volatile("global_load_async_to_lds_b128 %0, %1, %2"
                             :: "v"(laddr), "v"(off), "s"(sbase) : "memory");
            }
            asm volatile("s_wait_asynccnt 0x8" ::: "memory");
        } else {
            asm volatile("s_wait_asynccnt 0x0" ::: "memory");
        }
        __syncthreads();                          // stage st visible to all waves

        const float* bb = &zbuf[st & 1][0] + c0 + m;
        const int k0 = st * KT;
#pragma unroll
        for (int s = 0; s < KT / 4; ++s) {
            const int kb = s * 4 + koff;
            v2f a = {0.f, 0.f};
            if (m < HH) a = *reinterpret_cast<const v2f*>(arow + k0 + kb);
            v2f b;
            b.x = bb[kb * CC];                    // ds_load, conflict-free
            b.y = bb[(kb + 1) * CC];
            acc = wmma_f32(a, b, acc);
        }
        __syncthreads();                          // protect buffer reuse
    }
#pragma unroll
    for (int r = 0; r < 8; ++r) {
        const int mm = r + (lane >> 4) * 8;
        if (mm < HH) FEAT[(size_t)i * CAT + mm * CC + c0 + m] = acc[r];
    }
}

// ---------------- K4c: spatial aggregation alpha @ vp_global -----------------
__global__ void __launch_bounds__(32) k_aggr_sp(const float* __restrict__ ALPHA,
                                                const float* __restrict__ VPG,
                                                float* __restrict__ FEAT) {
    const int lane = threadIdx.x;
    const int ib = blockIdx.x * 16, h = blockIdx.y, n0 = blockIdx.z * 16;
    const int m = lane & 15;
    const int koff = (lane >> 4) * 2;
    const float* arow = ALPHA + (size_t)h * (LL * LL) + (size_t)(ib + m) * LL;
    const int nn = n0 + m;
    const bool valid = nn < (PV * 3);
    const float* bb = VPG + h * (PV * 3) + nn;
    v8f acc = {};
    for (int s = 0; s < LL / 4; ++s) {
        const int kb = s * 4 + koff;
        v2f a = *reinterpret_cast<const v2f*>(arow + kb);
        v2f b = {0.f, 0.f};
        if (valid) { b.x = bb[(size_t)kb * NP3]; b.y = bb[(size_t)(kb + 1) * NP3]; }
        acc = wmma_f32(a, b, acc);
    }
#pragma unroll
    for (int r = 0; r < 8; ++r) {
        const int mm = r + (lane >> 4) * 8;
        if (valid) FEAT[(size_t)(ib + mm) * CAT + (HH * CC + HH * VD) + h * (PV * 3) + nn] = acc[r];
    }
}

// ---------------- K5: global -> local (R^T (a - t)) in place in FEAT ---------
__global__ void k_g2l(const float* __restrict__ R, const float* __restrict__ t,
                      float* __restrict__ FEAT) {
    int gid = blockIdx.x * blockDim.x + threadIdx.x;
    if (gid >= LL * HH * PV) return;
    const int i = gid / (HH * PV);
    const int rem = gid % (HH * PV);
    const int h = rem / PV, p = rem % PV;
    float* f = FEAT + (size_t)i * CAT + (HH * CC + HH * VD) + h * (PV * 3) + p * 3;
    const float d0 = f[0] - t[i * 3 + 0];
    const float d1 = f[1] - t[i * 3 + 1];
    const float d2 = f[2] - t[i * 3 + 2];
    const float* Ri = R + i * 9;
    f[0] = Ri[0] * d0 + Ri[3] * d1 + Ri[6] * d2;
    f[1] = Ri[1] * d0 + Ri[4] * d1 + Ri[7] * d2;
    f[2] = Ri[2] * d0 + Ri[5] * d1 + Ri[8] * d2;
}

// ------------- generic WMMA GEMM: Y = act(A@W + bias)[mask] (+ Xadd) ---------
// A: LL x Kdim row-major (stride == Kdim); W: Kdim x Nstride row-major.
__global__ void __launch_bounds__(32) k_gemm(const float* __restrict__ A,
                                             const float* __restrict__ W,
                                             const float* __restrict__ bias,
                                             const float* __restrict__ Xadd,
                                             const unsigned char* __restrict__ mask,
                                             float* __restrict__ Y,
                                             int Kdim, int Nstride, int relu) {
    const int lane = threadIdx.x;
    const int ib = blockIdx.x * 16, n0 = blockIdx.y * 16;
    const int m = lane & 15;
    const int koff = (lane >> 4) * 2;
    const float* arow = A + (size_t)(ib + m) * Kdim;
    const float* wc = W + n0 + m;
    v8f acc = {};
    for (int s = 0; s < Kdim / 4; ++s) {
        const int kb = s * 4 + koff;
        v2f a = *reinterpret_cast<const v2f*>(arow + kb);
        v2f b;
        b.x = wc[(size_t)kb * Nstride];
        b.y = wc[(size_t)(kb + 1) * Nstride];
        acc = wmma_f32(a, b, acc);
    }
#pragma unroll
    for (int r = 0; r < 8; ++r) {
        const int mm = r + (lane >> 4) * 8;
        const int i = ib + mm, c = n0 + m;
        float v = acc[r] + (bias ? bias[c] : 0.f);
        if (relu) v = fmaxf(v, 0.f);
        if (mask && !mask[i]) v = 0.f;
        if (Xadd) v += Xadd[(size_t)i * Nstride + c];
        Y[(size_t)i * Nstride + c] = v;
    }
}

// ---------------- LayerNorm over last dim (128), optional residual add -------
__global__ void __launch_bounds__(32) k_ln(const float* __restrict__ A,
                                           const float* __restrict__ B,
                                           const float* __restrict__ sc,
                                           const float* __restrict__ of,
                                           float* __restrict__ out) {
    const int i = blockIdx.x, lane = threadIdx.x;
    float v[4];
    float sum = 0.f;
#pragma unroll
    for (int q = 0; q < 4; ++q) {
        const int c = lane + q * 32;
        float y = A[(size_t)i * FF + c];
        if (B) y += B[(size_t)i * FF + c];
        v[q] = y;
        sum += y;
    }
    const float mu = wsum(sum) * (1.f / FF);
    float var = 0.f;
#pragma unroll
    for (int q = 0; q < 4; ++q) { const float d = v[q] - mu; var = fmaf(d, d, var); }
    var = wsum(var) * (1.f / FF);
    const float inv = rsqrtf(var + 1e-5f);
#pragma unroll
    for (int q = 0; q < 4; ++q) {
        const int c = lane + q * 32;
        out[(size_t)i * FF + c] = (v[q] - mu) * inv * sc[c] + of[c];
    }
}

extern "C" void kernel_launch(void* const* d_in, const int* in_sizes, int n_in,
                              void* d_out, int out_size, void* d_ws, size_t ws_size,
                              hipStream_t stream) {
    const float* R    = (const float*)d_in[0];
    const float* t    = (const float*)d_in[1];
    const float* x    = (const float*)d_in[2];
    const float* z    = (const float*)d_in[3];
    const unsigned char* mask = (const unsigned char*)d_in[4];
    const float* Wq   = (const float*)d_in[5];
    const float* Wk   = (const float*)d_in[6];
    const float* Wpair= (const float*)d_in[7];
    const float* Wqp  = (const float*)d_in[8];
    const float* Wkp  = (const float*)d_in[9];
    const float* Wv   = (const float*)d_in[10];
    const float* Wvp  = (const float*)d_in[11];
    const float* coef = (const float*)d_in[12];
    const float* Wo   = (const float*)d_in[13];
    const float* bo   = (const float*)d_in[14];
    const float* ln1s = (const float*)d_in[15];
    const float* ln1o = (const float*)d_in[16];
    const float* W1   = (const float*)d_in[17];
    const float* b1   = (const float*)d_in[18];
    const float* W2   = (const float*)d_in[19];
    const float* b2   = (const float*)d_in[20];
    const float* W3   = (const float*)d_in[21];
    const float* b3   = (const float*)d_in[22];
    const float* ln2s = (const float*)d_in[23];
    const float* ln2o = (const float*)d_in[24];

    float* W = (float*)d_ws;
    float* Q    = W;                size_t off = (size_t)LL * NQK;   // 196608
    float* Kb   = W + off;          off += (size_t)LL * NQK;
    float* V    = W + off;          off += (size_t)LL * NQK;
    float* QPL  = W + off;          off += (size_t)LL * NP3;
    float* KPL  = W + off;          off += (size_t)LL * NP3;
    float* VPL  = W + off;          off += (size_t)LL * NP3;
    float* QPG  = W + off;          off += (size_t)LL * NP3;
    float* KPG  = W + off;          off += (size_t)LL * NP3;
    float* VPG  = W + off;          off += (size_t)LL * NP3;
    float* Q2   = W + off;          off += (size_t)LL * HH;
    float* K2   = W + off;          off += (size_t)LL * HH;
    float* LOG  = W + off;          off += (size_t)HH * LL * LL;     // 12.6 MB
    float* FEAT = W + off;          off += (size_t)LL * CAT;
    float* X1R  = W + off;          off += (size_t)LL * FF;
    float* X1   = W + off;          off += (size_t)LL * FF;
    float* H1   = W + off;          off += (size_t)LL * FF;
    float* H2   = W + off;          off += (size_t)LL * FF;
    float* MLP  = W + off;          off += (size_t)LL * FF;
    (void)ws_size; (void)in_sizes; (void)n_in; (void)out_size;

    k_proj<<<LL, 256, 0, stream>>>(x, Wq, Wk, Wv, Wqp, Wkp, Wvp,
                                   Q, Kb, V, QPL, KPL, VPL);
    k_l2g<<<(3 * LL * (NP3 / 3) + 255) / 256, 256, 0, stream>>>(R, t, QPL, KPL, VPL,
                                                                QPG, KPG, VPG);
    k_sumsq<<<(LL * HH + 255) / 256, 256, 0, stream>>>(QPG, KPG, Q2, K2);

    k_pair<<<(LL * LL) / 16, 32, 0, stream>>>(z, Wpair, LOG);
    k_logits<<<dim3(LL / 16, LL / 16, HH), 32, 0, stream>>>(Q, Kb, QPG, KPG, Q2, K2,
                                                            coef, mask, LOG);
    k_softmax<<<dim3(LL, HH), 32, 0, stream>>>(mask, LOG);

    k_aggr_node<<<dim3(LL / 16, HH, VD / 16), 32, 0, stream>>>(LOG, V, FEAT);
    k_aggr_pair<<<LL, 128, 0, stream>>>(LOG, z, FEAT);
    k_aggr_sp<<<dim3(LL / 16, HH, 2), 32, 0, stream>>>(LOG, VPG, FEAT);
    k_g2l<<<(LL * HH * PV + 255) / 256, 256, 0, stream>>>(R, t, FEAT);

    // x1_raw = x + where(mask, FEAT@Wo + bo, 0)
    k_gemm<<<dim3(LL / 16, FF / 16), 32, 0, stream>>>(FEAT, Wo, bo, x, mask, X1R,
                                                      CAT, FF, 0);
    k_ln<<<LL, 32, 0, stream>>>(X1R, nullptr, ln1s, ln1o, X1);

    k_gemm<<<dim3(LL / 16, FF / 16), 32, 0, stream>>>(X1, W1, b1, nullptr, nullptr, H1,
                                                      FF, FF, 1);
    k_gemm<<<dim3(LL / 16, FF / 16), 32, 0, stream>>>(H1, W2, b2, nullptr, nullptr, H2,
                                                      FF, FF, 1);
    k_gemm<<<dim3(LL / 16, FF / 16), 32, 0, stream>>>(H2, W3, b3, nullptr, nullptr, MLP,
                                                      FF, FF, 0);
    k_ln<<<LL, 32, 0, stream>>>(X1, MLP, ln2s, ln2o, (float*)d_out);
}